// Model_8899172237430
// MI455X (gfx1250) — compile-verified
//
#include <hip/hip_runtime.h>
#include <hip/hip_bf16.h>

// Problem constants (match reference)
#define BDIM 32768
#define MDIM 3
#define HDIM 768
#define KDIM 512
#define SK_EPS_F 0.03f
#define SK_ITERS 100

typedef __attribute__((ext_vector_type(2))) float v2f;
typedef __attribute__((ext_vector_type(8))) float v8f;

// ---- ordered-uint tricks for global float min/max via integer atomics ----
__device__ __forceinline__ unsigned f2ord(float f) {
    unsigned u = __float_as_uint(f);
    return (u & 0x80000000u) ? ~u : (u | 0x80000000u);
}
__device__ __forceinline__ float ord2f(unsigned o) {
    unsigned u = (o & 0x80000000u) ? (o & 0x7FFFFFFFu) : ~o;
    return __uint_as_float(u);
}

// ws layout (32-bit words):
//  [0] ordmax(prob)  [1] ordmin(prob)  [2] middle  [3] 1/(amp*eps)
//  [4 .. 4+512)   S_k   (column sums of E)
//  [516 .. 1028)  ln f_k (per-row sinkhorn log-scale)

// ---------------------------------------------------------------------------
// init: zero S_k, seed min/max atomics
__global__ void init_kernel(unsigned* __restrict__ ws_u, float* __restrict__ Sk) {
    int t = threadIdx.x;                 // 512 threads
    Sk[t] = 0.0f;
    if (t == 0) { ws_u[0] = 0u; ws_u[1] = 0xFFFFFFFFu; }
}

// ---------------------------------------------------------------------------
// Fused GEMM: code_logits[b,m,k] = sum_h dec[b,m,h]*cent[m,k,h]  (fp32 WMMA)
// Block = 4 waves sharing the SAME 16 rows of A (single HBM pass over dec);
// wave w covers cols [128w, 128w+128) via 8 C frags -> block = 16 x 512 tile.
// grid = (B/16, M)
__global__ __launch_bounds__(128) void gemm_kernel(
    const float* __restrict__ dec, const float* __restrict__ cent,
    float* __restrict__ out_logits, float* __restrict__ out_prob,
    float* __restrict__ out_ret, unsigned* __restrict__ ws_u)
{
    const int m    = blockIdx.y;
    const int wave = threadIdx.x >> 5;
    const int lane = threadIdx.x & 31;
    const int half = lane >> 4;      // 0: K={0,1}, 1: K={2,3} within each x4 step
    const int l16  = lane & 15;
    const int rowBase = blockIdx.x * 16;
    const int colBase = wave * 128;

    // A fragment source: row (rowBase+l16) of dec[:,m,:], starting at h+2*half.
    // All 4 waves issue identical A addresses -> WGP$/L2 hits, one HBM fetch.
    const float* Aptr = dec + ((size_t)(rowBase + l16) * MDIM + m) * HDIM + 2 * half;
    const float* cm   = cent + (size_t)m * KDIM * HDIM;
    const float* Bp0  = cm + (size_t)(colBase + l16) * HDIM + 2 * half;

    v8f acc[8] = {};

    for (int h = 0; h < HDIM; h += 4) {
        v2f a = *(const v2f*)(Aptr + h);
#pragma unroll
        for (int j = 0; j < 8; ++j) {
            v2f b = *(const v2f*)(Bp0 + (size_t)j * 16 * HDIM + h);
            // D = A(16x4,f32) * B(4x16,f32) + C(16x16,f32)
            acc[j] = __builtin_amdgcn_wmma_f32_16x16x4_f32(
                false, a, false, b, (short)0, acc[j], false, false);
        }
    }

    // Writeback per ISA C layout: VGPR i -> M = i + 8*half, N = l16
    float lmin = 3.402823e38f, lmax = -3.402823e38f;
#pragma unroll
    for (int j = 0; j < 8; ++j) {
        const int col = colBase + j * 16 + l16;
#pragma unroll
        for (int i = 0; i < 8; ++i) {
            const int row = rowBase + i + 8 * half;
            const float v = acc[j][i];
            out_logits[((size_t)row * MDIM + m) * KDIM + col] = v;
            if (m == 2) {
                out_prob[(size_t)row * KDIM + col] = v;
                lmin = fminf(lmin, v); lmax = fmaxf(lmax, v);
            } else {
                out_ret[((size_t)row * 2 + m) * KDIM + col] = v;
            }
        }
    }

    if (m == 2) {
        __shared__ unsigned smax[128], smin[128];
        smax[threadIdx.x] = f2ord(lmax);
        smin[threadIdx.x] = f2ord(lmin);
        __syncthreads();
        for (int s = 64; s > 0; s >>= 1) {
            if (threadIdx.x < s) {
                smax[threadIdx.x] = max(smax[threadIdx.x], smax[threadIdx.x + s]);
                smin[threadIdx.x] = min(smin[threadIdx.x], smin[threadIdx.x + s]);
            }
            __syncthreads();
        }
        if (threadIdx.x == 0) {
            atomicMax(&ws_u[0], smax[0]);
            atomicMin(&ws_u[1], smin[0]);
        }
    }
}

// ---------------------------------------------------------------------------
// scalar: middle, 1/(amplitude*eps).  distances d = -prob.
__global__ void minmax_finalize(const unsigned* __restrict__ ws_u, float* __restrict__ ws_f) {
    float maxp = ord2f(ws_u[0]);
    float minp = ord2f(ws_u[1]);
    float mx = -minp, mn = -maxp;                // max/min of d = -prob
    float middle = 0.5f * (mx + mn);
    float amp    = mx - middle + 1e-5f;
    ws_f[2] = middle;
    ws_f[3] = 1.0f / (amp * SK_EPS_F);
}

// ---------------------------------------------------------------------------
// S_k = sum_b exp((prob[b,k]+middle)*inv).  grid=256 blocks x 512 threads,
// thread t owns column k=t (coalesced row-major reads), 128 rows per block.
__global__ __launch_bounds__(512) void colsum_kernel(
    const float* __restrict__ prob, const float* __restrict__ ws_f,
    float* __restrict__ Sk)
{
    const int k = threadIdx.x;
    const float middle = ws_f[2], inv = ws_f[3];
    const int r0 = blockIdx.x * 128;
    float acc = 0.0f;
    for (int r = 0; r < 128; ++r)
        acc += expf((prob[(size_t)(r0 + r) * KDIM + k] + middle) * inv);
    atomicAdd(&Sk[k], acc);
}

// ---------------------------------------------------------------------------
// Faithful 100-iter Sinkhorn collapsed to per-row scale f_k (Q = E*f_k at all
// times since only row + global rescales occur).  Includes the 1e-5 clips.
__global__ __launch_bounds__(512) void sinkhorn_kernel(
    const float* __restrict__ Sk, float* __restrict__ lnf)
{
    const int t = threadIdx.x;
    __shared__ float red[512];
    const float S = Sk[t];

    red[t] = S; __syncthreads();
    for (int s = 256; s > 0; s >>= 1) { if (t < s) red[t] += red[t + s]; __syncthreads(); }
    float f = 1.0f / fmaxf(red[0], 1e-5f);       // initial global normalization
    __syncthreads();

    for (int it = 0; it < SK_ITERS; ++it) {
        float row = fmaxf(S * f, 1e-5f);         // rows = clip(sum_b Q[k,:])
        f = f / row * (1.0f / KDIM);             // Q /= rows / K   (1/512 exact)
        red[t] = S * f; __syncthreads();
        for (int s = 256; s > 0; s >>= 1) { if (t < s) red[t] += red[t + s]; __syncthreads(); }
        float tot = fmaxf(red[0], 1e-5f);
        __syncthreads();
        f = f / tot * (1.0f / BDIM);             // Q /= total / B  (1/32768 exact)
    }
    lnf[t] = logf(f);
}

// ---------------------------------------------------------------------------
// argmax_k (prob[b,k]*inv + lnf[k])  ->  code; gather code_emb_last[code].
// grid = B blocks x 256 threads.
__global__ __launch_bounds__(256) void argmax_gather_kernel(
    const float* __restrict__ prob, const float* __restrict__ ws_f,
    const float* __restrict__ lnf, const float* __restrict__ emb,
    float* __restrict__ out_codes, float* __restrict__ out_quant)
{
    const int b = blockIdx.x;
    const int t = threadIdx.x;
    const float inv = ws_f[3];
    __shared__ float bv[256]; __shared__ int bi[256]; __shared__ int codeSh;

    float best = -3.402823e38f; int bidx = 0;
#pragma unroll
    for (int j = 0; j < 2; ++j) {
        int k = t + j * 256;
        float sc = prob[(size_t)b * KDIM + k] * inv + lnf[k];
        if (sc > best) { best = sc; bidx = k; }
    }
    bv[t] = best; bi[t] = bidx; __syncthreads();
    for (int s = 128; s > 0; s >>= 1) {
        if (t < s) {
            // jnp.argmax: first (lowest-index) max wins ties
            if (bv[t + s] > bv[t] || (bv[t + s] == bv[t] && bi[t + s] < bi[t])) {
                bv[t] = bv[t + s]; bi[t] = bi[t + s];
            }
        }
        __syncthreads();
    }
    if (t == 0) { codeSh = bi[0]; out_codes[b] = (float)bi[0]; }
    __syncthreads();
    const int code = codeSh;
    const float4* src = (const float4*)(emb + (size_t)code * HDIM);
    float4*       dst = (float4*)(out_quant + (size_t)b * HDIM);
    if (t < HDIM / 4) dst[t] = src[t];           // 192 float4s
}

// ---------------------------------------------------------------------------
// continuous_embeds = decoder_outputs[:, 2, :]  (strided float4 copy)
__global__ void cont_copy_kernel(const float4* __restrict__ dec4, float4* __restrict__ out4) {
    int idx = blockIdx.x * blockDim.x + threadIdx.x;    // 0 .. B*192-1
    int b = idx / (HDIM / 4), w = idx % (HDIM / 4);
    out4[idx] = dec4[(size_t)(b * MDIM + 2) * (HDIM / 4) + w];
}

// ---------------------------------------------------------------------------
extern "C" void kernel_launch(void* const* d_in, const int* in_sizes, int n_in,
                              void* d_out, int out_size, void* d_ws, size_t ws_size,
                              hipStream_t stream) {
    const float* dec  = (const float*)d_in[0];   // (B, 3, 768)
    const float* cent = (const float*)d_in[1];   // (3, 512, 768)
    const float* emb  = (const float*)d_in[2];   // (512, 768)
    float* out = (float*)d_out;

    // Outputs concatenated flat in return order
    const size_t n_logits = (size_t)BDIM * MDIM * KDIM;   // 50331648
    const size_t n_dense  = (size_t)BDIM * MDIM * HDIM;   // 75497472
    const size_t n_cont   = (size_t)BDIM * HDIM;          // 25165824
    float* out_logits = out;
    float* out_dense  = out_logits + n_logits;
    float* out_cont   = out_dense  + n_dense;
    float* out_quant  = out_cont   + n_cont;
    float* out_codes  = out_quant  + n_cont;              // codes stored as float
    float* out_prob   = out_codes  + BDIM;
    float* out_ret    = out_prob   + (size_t)BDIM * KDIM;

    unsigned* ws_u = (unsigned*)d_ws;
    float*    ws_f = (float*)d_ws;
    float*    Sk   = ws_f + 4;
    float*    lnf  = Sk + KDIM;

    init_kernel<<<1, 512, 0, stream>>>(ws_u, Sk);

    dim3 g(BDIM / 16, MDIM);                       // 2048 x 3 blocks, 16x512 tile each
    gemm_kernel<<<g, 128, 0, stream>>>(dec, cent, out_logits, out_prob, out_ret, ws_u);

    minmax_finalize<<<1, 1, 0, stream>>>(ws_u, ws_f);
    colsum_kernel<<<256, 512, 0, stream>>>(out_prob, ws_f, Sk);
    sinkhorn_kernel<<<1, 512, 0, stream>>>(Sk, lnf);
    argmax_gather_kernel<<<BDIM, 256, 0, stream>>>(out_prob, ws_f, lnf, emb,
                                                   out_codes, out_quant);
    cont_copy_kernel<<<(BDIM * (HDIM / 4)) / 256, 256, 0, stream>>>(
        (const float4*)dec, (float4*)out_cont);

    hipMemcpyAsync(out_dense, dec, n_dense * sizeof(float),
                   hipMemcpyDeviceToDevice, stream);
}